// HybridEstimatorQNN_65481071405119
// MI455X (gfx1250) — compile-verified
//
#include <hip/hip_runtime.h>
#include <math.h>

// MI455X fused kernel for the "HybridEstimatorQNN":
// quantum_features collapses analytically to qf = cos*cos products
// (product state, CRX(0)=I, RZ diagonal), so the whole net is a tiny
// fused MLP: out = tanh([x,qf] @ W1 + b1) @ W2 + b2, K=16, H=256.
// The 16x16 (batch-tile x K) GEMM against W1 runs on exact-fp32
// V_WMMA_F32_16X16X4_F32 (4 chained K-steps x 16 N-tiles = 64 wmma/wave).
// Transcendentals use branch-free v_exp_f32 / v_cos_f32 hardware paths so
// no EXEC-divergent libm code interleaves with the WMMA stream.

typedef __attribute__((ext_vector_type(2))) float v2f;
typedef __attribute__((ext_vector_type(8))) float v8f;

#define DIN  6
#define HDIM 256

// branch-free tanh: sign(x) * (1 - e^{-2|x|}) / (1 + e^{-2|x|})
// negative exponent never overflows; underflow -> tanh = +/-1 exactly.
__device__ __forceinline__ float tanh_fast(float x) {
  float ax = __builtin_fabsf(x);
  float t  = __expf(-2.0f * ax);                      // v_exp_f32 path, t in (0,1]
  float y  = (1.0f - t) * __builtin_amdgcn_rcpf(1.0f + t);  // v_rcp_f32
  return __builtin_copysignf(y, x);                   // bitop
}

// combined[c] for c in [0,16): cols 0..5 = x, cols 6..15 alternate f0,f1
template <int C>
__device__ __forceinline__ float ccol(const float xv[6], float f0, float f1) {
  if constexpr (C < DIN) return xv[C];
  else if constexpr ((C & 1) == 0) return f0;   // cols 6,8,10,12,14 -> cos(x0)cos(x1)
  else return f1;                               // cols 7,9,11,13,15 -> cos(x3)cos(x4)
}

// A fragment for K-chunk KC of V_WMMA_F32_16X16X4_F32:
// lanes 0-15: VGPR0=K(4KC+0), VGPR1=K(4KC+1); lanes 16-31: K(4KC+2), K(4KC+3)
template <int KC>
__device__ __forceinline__ v2f make_a(const float xv[6], float f0, float f1, bool hi) {
  float a0lo = ccol<4 * KC + 0>(xv, f0, f1);
  float a1lo = ccol<4 * KC + 1>(xv, f0, f1);
  float a0hi = ccol<4 * KC + 2>(xv, f0, f1);
  float a1hi = ccol<4 * KC + 3>(xv, f0, f1);
  v2f a;
  a.x = hi ? a0hi : a0lo;
  a.y = hi ? a1hi : a1lo;
  return a;
}

__global__ __launch_bounds__(256) void qnn_fused_kernel(
    const float* __restrict__ x,   // [B, 6]
    const float* __restrict__ W1,  // [16, 256] row-major
    const float* __restrict__ b1,  // [256]
    const float* __restrict__ W2,  // [256]  (H x 1)
    const float* __restrict__ b2,  // [1]
    float* __restrict__ out,       // [B]
    int B, int ntiles)
{
  const int lane = threadIdx.x & 31;
  const int wave = threadIdx.x >> 5;
  const int tile = blockIdx.x * (blockDim.x >> 5) + wave;  // 16 samples per wave
  if (tile >= ntiles) return;  // wave-uniform: EXEC stays all-1s for live waves

  const int m  = lane & 15;
  const bool hi = lane >= 16;

  int row = tile * 16 + m;
  int rowc = row < B ? row : (B - 1);  // clamp loads for a (non-occurring) ragged tile
  const float* xp = x + (long)rowc * DIN;

  float xv[6];
#pragma unroll
  for (int i = 0; i < DIN; ++i) xv[i] = xp[i];

  // analytic <Z_q>: even q -> cos(x0)cos(x1), odd q -> cos(x3)cos(x4)
  // inputs ~N(0,1): hardware v_cos_f32 is exact enough
  float f0 = __cosf(xv[0]) * __cosf(xv[1]);
  float f1 = __cosf(xv[3]) * __cosf(xv[4]);

  // A fragments are N-tile invariant: build once
  v2f A0 = make_a<0>(xv, f0, f1, hi);
  v2f A1 = make_a<1>(xv, f0, f1, hi);
  v2f A2 = make_a<2>(xv, f0, f1, hi);
  v2f A3 = make_a<3>(xv, f0, f1, hi);

  const int krow0 = hi ? 2 : 0;

  float acc[8];
#pragma unroll
  for (int r = 0; r < 8; ++r) acc[r] = 0.0f;

#pragma unroll
  for (int nt = 0; nt < HDIM / 16; ++nt) {
    const int n = nt * 16 + m;  // hidden-unit column this lane owns

    v8f C = {};
#pragma unroll
    for (int kc = 0; kc < 4; ++kc) {
      v2f Bf;
      Bf.x = W1[(4 * kc + krow0 + 0) * HDIM + n];
      Bf.y = W1[(4 * kc + krow0 + 1) * HDIM + n];
      v2f Af = (kc == 0) ? A0 : (kc == 1) ? A1 : (kc == 2) ? A2 : A3;
      // D = A(16x4) x B(4x16) + C, exact fp32
      C = __builtin_amdgcn_wmma_f32_16x16x4_f32(
          /*neg_a=*/false, Af, /*neg_b=*/false, Bf,
          /*c_mod=*/(short)0, C, /*reuse_a=*/false, /*reuse_b=*/false);
    }

    const float b1n = b1[n];
    const float w2n = W2[n];
    // C VGPR r: lanes 0-15 -> sample m=r, lanes 16-31 -> sample m=8+r, N=n
#pragma unroll
    for (int r = 0; r < 8; ++r) {
      acc[r] = __builtin_fmaf(tanh_fast(C[r] + b1n), w2n, acc[r]);
    }
  }

  const float bias2 = b2[0];
  const int base = tile * 16;

  // reduce 16 hidden-column partials within each 16-lane half (wave32)
#pragma unroll
  for (int r = 0; r < 8; ++r) {
    float v = acc[r];
    v += __shfl_xor(v, 1, 32);
    v += __shfl_xor(v, 2, 32);
    v += __shfl_xor(v, 4, 32);
    v += __shfl_xor(v, 8, 32);
    // lanes 0-15 now hold sample base+r; lanes 16-31 hold sample base+8+r
    if (lane == r && base + r < B)            out[base + r]     = v + bias2;
    if (lane == 16 + r && base + 8 + r < B)   out[base + 8 + r] = v + bias2;
  }
}

extern "C" void kernel_launch(void* const* d_in, const int* in_sizes, int n_in,
                              void* d_out, int out_size, void* d_ws, size_t ws_size,
                              hipStream_t stream) {
  const float* x  = (const float*)d_in[0];  // [B,6]
  const float* W1 = (const float*)d_in[1];  // [16,256]
  const float* b1 = (const float*)d_in[2];  // [256]
  const float* W2 = (const float*)d_in[3];  // [256,1]
  const float* b2 = (const float*)d_in[4];  // [1]
  float* out = (float*)d_out;

  const int B = in_sizes[0] / DIN;           // 16384
  const int ntiles = (B + 15) / 16;          // 1024 waves of work
  const int wavesPerBlock = 8;               // 256 threads
  const int blocks = (ntiles + wavesPerBlock - 1) / wavesPerBlock;

  qnn_fused_kernel<<<blocks, wavesPerBlock * 32, 0, stream>>>(
      x, W1, b1, W2, b2, out, B, ntiles);
}